// SumPooling_910533067557
// MI455X (gfx1250) — compile-verified
//
#include <hip/hip_runtime.h>

#define D_FEAT 128
#define GROUPS 16              // 16-row groups per block chunk
#define CHUNK  (GROUPS * 16)   // 256 rows per block

typedef __attribute__((ext_vector_type(2))) float v2f;
typedef __attribute__((ext_vector_type(8))) float v8f;

// Flush the 16(seg) x 16(feat) accumulator window to global with sparse atomics.
// C/D layout: VGPR v -> M = v (lanes 0-15) / v+8 (lanes 16-31); N = lane & 15.
__device__ __forceinline__ void flush_acc(float* __restrict__ out, v8f& c,
                                          int win, int lane, int col_base) {
  const int m_hi = (lane >> 4) << 3;
  const int col  = col_base + (lane & 15);
#pragma unroll
  for (int v = 0; v < 8; ++v) {
    float val = c[v];
    if (val != 0.0f) {
      int seg = win + v + m_hi;
      atomicAdd(&out[(size_t)seg * D_FEAT + col], val);
    }
    c[v] = 0.0f;
  }
}

__global__ __launch_bounds__(256) void seg_sum_wmma(
    const float* __restrict__ x, const long long* __restrict__ idx,
    float* __restrict__ out, int n) {
  const int lane     = threadIdx.x & 31;
  const int wave     = threadIdx.x >> 5;
  const int col_base = wave * 16;        // 8 waves x 16 cols = 128 features
  const int m        = lane & 15;
  const int kloc     = (lane >> 4) << 1; // A/B f32 layout: lanes 0-15 K=0,1; 16-31 K=2,3
  const int chunk0   = blockIdx.x * CHUNK;

  v8f c;
#pragma unroll
  for (int v = 0; v < 8; ++v) c[v] = 0.0f;
  int win = -1;                          // current 16-wide segment window base

  for (int g = 0; g < GROUPS; ++g) {
    const int rb = chunk0 + g * 16;
    if (rb >= n) break;                                  // wave-uniform
    const int  lastlane = min(15, n - 1 - rb);
    const int  myrow    = rb + m;
    const bool valid    = myrow < n;
    // Clamped, unconditional index load (always in-bounds; A masks invalid rows).
    const int  myseg    = (int)idx[min(myrow, n - 1)];
    const int  gmin     = __shfl(myseg, 0);
    const int  gmax     = __shfl(myseg, lastlane);

    if (win >= 0 && gmax >= win + 16) {                  // window overflow
      flush_acc(out, c, win, lane, col_base);
      win = -1;
    }
    if (win < 0) {
      if (gmax - gmin >= 16) {
        // Degenerate group spanning >=16 segments: scalar atomic fallback.
        for (int t = 0; t < 8; ++t) {
          int lin = lane + 32 * t;       // 0..255 = 16 rows x 16 cols
          int ri  = lin >> 4;
          int cc  = lin & 15;
          int row = rb + ri;
          if (row < n) {
            int seg = (int)idx[row];
            atomicAdd(&out[(size_t)seg * D_FEAT + col_base + cc],
                      x[(size_t)row * D_FEAT + col_base + cc]);
          }
        }
        continue;
      }
      win = gmin;                        // sorted => gmin >= old window base
    }

    // One-hot column for this lane's row (-1 => contributes zero via A).
    const int rel = valid ? (myseg - win) : -1;

    // Prefetch next group's tile (global_prefetch_b8; speculative, drop-on-fault).
    if (rb + 16 + m < n)
      __builtin_prefetch(&x[(size_t)(rb + 16 + m) * D_FEAT + col_base], 0, 1);

    // ---- Gather B: 16 rows x 16 cols of x, 2 f32 per lane per WMMA ----
    v2f a[4], b[4];
    if (rb + 16 <= n) {
      // Full group (all but the grid's last group): single base address,
      // immediate byte offsets kk*2048 / kk*2048+512 folded into the loads.
      const float* xg = x + (size_t)(rb + kloc) * D_FEAT + col_base + m;
#pragma unroll
      for (int kk = 0; kk < 4; ++kk) {
        b[kk].x = xg[(size_t)(kk * 4) * D_FEAT];
        b[kk].y = xg[(size_t)(kk * 4 + 1) * D_FEAT];
      }
    } else {
      // Tail group: clamp rows to n-1 (A's one-hot zeroes their contribution).
#pragma unroll
      for (int kk = 0; kk < 4; ++kk) {
        int r0 = min(rb + kk * 4 + kloc, n - 1);
        int r1 = min(rb + kk * 4 + kloc + 1, n - 1);
        b[kk].x = x[(size_t)r0 * D_FEAT + col_base + m];
        b[kk].y = x[(size_t)r1 * D_FEAT + col_base + m];
      }
    }

    // ---- Build one-hot A from shuffled per-row window offsets ----
#pragma unroll
    for (int kk = 0; kk < 4; ++kk) {
      int rel0 = __shfl(rel, kk * 4 + kloc);
      int rel1 = __shfl(rel, kk * 4 + kloc + 1);
      a[kk].x = (rel0 == m) ? 1.0f : 0.0f;
      a[kk].y = (rel1 == m) ? 1.0f : 0.0f;
    }

    // ---- Four 16x16x4 f32 WMMAs: C += one_hot(A) x B (exact f32) ----
#pragma unroll
    for (int kk = 0; kk < 4; ++kk)
      c = __builtin_amdgcn_wmma_f32_16x16x4_f32(false, a[kk], false, b[kk],
                                                (short)0, c, false, false);
  }
  if (win >= 0) flush_acc(out, c, win, lane, col_base);
}

extern "C" void kernel_launch(void* const* d_in, const int* in_sizes, int n_in,
                              void* d_out, int out_size, void* d_ws, size_t ws_size,
                              hipStream_t stream) {
  const float*     x   = (const float*)d_in[0];
  const long long* idx = (const long long*)d_in[1];  // reference dtype: int64
  float*           out = (float*)d_out;
  const int n = in_sizes[1];                         // N_ELEMENTS (index count)

  // d_out is poisoned by the harness; we accumulate with atomics, so zero it.
  hipMemsetAsync(d_out, 0, (size_t)out_size * sizeof(float), stream);

  const int nblocks = (n + CHUNK - 1) / CHUNK;
  seg_sum_wmma<<<nblocks, 256, 0, stream>>>(x, idx, out, n);
}